// AttentionLayerWithMask_36532991819812
// MI455X (gfx1250) — compile-verified
//
#include <hip/hip_runtime.h>
#include <hip/hip_bf16.h>
#include <math.h>

// Problem constants (from reference): B=1024, S=200, D=512, O=512, fp32 everywhere.
//
// Algebraic rewrite (exact, softmax-invariant):
//   scores[b,s] = (Wk^T q_b) . x[b,s]            (q.b_k const per row -> dropped)
//   out[b]      = tanh( Wv . (sum_s attn_s x[b,s]) + b_v )
// -> three [1024,512]x[512,512] fp32 WMMA GEMMs + two HBM-bound streaming passes.

typedef __attribute__((ext_vector_type(2))) float v2f;
typedef __attribute__((ext_vector_type(8))) float v8f;

// ---------------------------------------------------------------------------
// Kernel 1: sqlen[b] = sum(mask[b,:]); last[b,:] = x[b, sqlen-1, :]
// ---------------------------------------------------------------------------
__global__ __launch_bounds__(256)
void prep_kernel(const float* __restrict__ x, const int* __restrict__ mask,
                 int* __restrict__ sqlen, float* __restrict__ last,
                 int S, int D) {
    const int b = blockIdx.x;
    const int tid = threadIdx.x;
    __shared__ int red[256];
    red[tid] = (tid < S) ? mask[(size_t)b * S + tid] : 0;
    __syncthreads();
    #pragma unroll
    for (int s = 128; s > 0; s >>= 1) {
        if (tid < s) red[tid] += red[tid + s];
        __syncthreads();
    }
    const int sq = red[0];
    if (tid == 0) sqlen[b] = sq;
    const float* src = x + ((size_t)b * S + (sq - 1)) * D;
    for (int d = tid; d < D; d += 256)
        last[(size_t)b * D + d] = src[d];
}

// ---------------------------------------------------------------------------
// WMMA fp32 GEMM: Out[M,N] = act(A[M,K] @ op(W) + bias)
//   TRANSB=1: op(W)=W^T with W stored [N,K]   (projection with weight [O,D])
//   TRANSB=0: op(W)=W   with W stored [K,N]
// One wave computes a 16(M) x 64(N) strip: 4 accumulators, A-fragment reused.
// V_WMMA_F32_16X16X4_F32 layouts per ISA 7.12.2:
//   A 16x4:  lane<16 -> (M=lane, K={0,1}); lane>=16 -> (M=lane-16, K={2,3})
//   B 4x16:  mirrored on N;  C/D: VGPR r -> row r (lanes 0-15) / r+8 (16-31)
// ---------------------------------------------------------------------------
template<bool TRANSB, bool HAS_BIAS, bool TANH_ACT>
__global__ __launch_bounds__(128)
void gemm_wmma_f32(const float* __restrict__ A, const float* __restrict__ W,
                   const float* __restrict__ bias, float* __restrict__ Out,
                   int M, int N, int K) {
    const int lane = threadIdx.x & 31;
    const int wid  = blockIdx.x * 4 + (threadIdx.x >> 5);
    const int nTilesN = N >> 6;                 // 64-wide N strips
    const int m0 = (wid / nTilesN) << 4;
    const int n0 = (wid % nTilesN) << 6;
    if (m0 >= M) return;

    const int r16 = lane & 15;                  // M-row (A) / N-col (B,C)
    const int kh  = (lane >> 4) << 1;           // K sub-offset: 0 or 2

    v8f c[4] = {v8f{}, v8f{}, v8f{}, v8f{}};
    const float* arow = A + (size_t)(m0 + r16) * K;

    for (int k0 = 0; k0 < K; k0 += 4) {
        const int ka = k0 + kh;
        const v2f a = *(const v2f*)(arow + ka);
        v2f bfrag[4];
        if (TRANSB) {
            #pragma unroll
            for (int t = 0; t < 4; ++t)
                bfrag[t] = *(const v2f*)(W + (size_t)(n0 + t * 16 + r16) * K + ka);
        } else {
            const float* w0 = W + (size_t)ka * N + n0 + r16;
            #pragma unroll
            for (int t = 0; t < 4; ++t) {
                bfrag[t].x = w0[t * 16];
                bfrag[t].y = w0[t * 16 + N];
            }
        }
        #pragma unroll
        for (int t = 0; t < 4; ++t)
            c[t] = __builtin_amdgcn_wmma_f32_16x16x4_f32(
                false, a, false, bfrag[t], (short)0, c[t], false, false);
    }

    const int rowBase = m0 + ((lane >> 4) << 3);
    #pragma unroll
    for (int t = 0; t < 4; ++t) {
        const int col = n0 + t * 16 + r16;
        const float bv = HAS_BIAS ? bias[col] : 0.0f;
        #pragma unroll
        for (int r = 0; r < 8; ++r) {
            float v = c[t][r] + bv;
            if (TANH_ACT) v = tanhf(v);
            Out[(size_t)(rowBase + r) * N + col] = v;
        }
    }
}

// ---------------------------------------------------------------------------
// Kernel: scores + masked softmax, fused per batch row.
//   scores[b,s] = T[b,:] . x[b,s,:]  for s < sqlen-1, then softmax -> attn.
// 8 waves/block; each wave dots one token row (2KB) per iteration.
// ---------------------------------------------------------------------------
__global__ __launch_bounds__(256)
void scores_softmax_kernel(const float* __restrict__ x, const float* __restrict__ T,
                           const int* __restrict__ sqlen, float* __restrict__ attn,
                           int S, int D) {
    const int b = blockIdx.x;
    const int tid = threadIdx.x;
    const int lane = tid & 31;
    const int wave = tid >> 5;
    __shared__ float tv[512];
    __shared__ float sc[256];
    __shared__ float red[256];

    for (int d = tid; d < D; d += 256) tv[d] = T[(size_t)b * D + d];
    const int send = sqlen[b] - 1;              // #valid keys, >= 1
    __syncthreads();

    for (int s = wave; s < send; s += 8) {
        const float* xp = x + ((size_t)b * S + s) * D;
        float acc = 0.0f;
        #pragma unroll
        for (int j = 0; j < 4; ++j) {
            const int d = j * 128 + lane * 4;   // coalesced float4 across wave
            const float4 xv = *(const float4*)(xp + d);
            acc += xv.x * tv[d] + xv.y * tv[d + 1] + xv.z * tv[d + 2] + xv.w * tv[d + 3];
        }
        #pragma unroll
        for (int off = 16; off > 0; off >>= 1)
            acc += __shfl_xor(acc, off, 32);
        if (lane == 0) sc[s] = acc;
    }
    __syncthreads();

    // block softmax over sc[0..send)
    red[tid] = (tid < send) ? sc[tid] : -__builtin_inff();
    __syncthreads();
    #pragma unroll
    for (int s = 128; s > 0; s >>= 1) {
        if (tid < s) red[tid] = fmaxf(red[tid], red[tid + s]);
        __syncthreads();
    }
    const float mx = red[0];
    __syncthreads();
    const float e = (tid < send) ? __expf(sc[tid] - mx) : 0.0f;
    red[tid] = e;
    __syncthreads();
    #pragma unroll
    for (int s = 128; s > 0; s >>= 1) {
        if (tid < s) red[tid] += red[tid + s];
        __syncthreads();
    }
    const float inv = 1.0f / red[0];
    if (tid < S) attn[(size_t)b * S + tid] = e * inv;
}

// ---------------------------------------------------------------------------
// Kernel: xbar[b,d] = sum_{s < sqlen-1} attn[b,s] * x[b,s,d]
// ---------------------------------------------------------------------------
__global__ __launch_bounds__(256)
void xbar_kernel(const float* __restrict__ x, const float* __restrict__ attn,
                 const int* __restrict__ sqlen, float* __restrict__ xbar,
                 int S, int D) {
    const int b = blockIdx.x;
    const int tid = threadIdx.x;
    __shared__ float a[256];
    const int send = sqlen[b] - 1;
    a[tid] = (tid < send && tid < S) ? attn[(size_t)b * S + tid] : 0.0f;
    __syncthreads();

    const float* xb = x + (size_t)b * S * D;
    float acc0 = 0.0f, acc1 = 0.0f;
    const int d0 = tid, d1 = tid + 256;
    for (int s = 0; s < send; ++s) {
        const float w = a[s];
        acc0 += w * xb[(size_t)s * D + d0];
        acc1 += w * xb[(size_t)s * D + d1];
    }
    xbar[(size_t)b * D + d0] = acc0;
    xbar[(size_t)b * D + d1] = acc1;
}

// ---------------------------------------------------------------------------
extern "C" void kernel_launch(void* const* d_in, const int* in_sizes, int n_in,
                              void* d_out, int out_size, void* d_ws, size_t ws_size,
                              hipStream_t stream) {
    (void)in_sizes; (void)n_in; (void)out_size; (void)ws_size;
    const int B = 1024, S = 200, D = 512, O = 512;

    const float* x    = (const float*)d_in[0];
    const int*   mask = (const int*)  d_in[1];
    const float* Wq_w = (const float*)d_in[2];
    const float* Wq_b = (const float*)d_in[3];
    const float* Wk_w = (const float*)d_in[4];
    // d_in[5] (Wk_b) intentionally unused: constant per row -> softmax-invariant.
    const float* Wv_w = (const float*)d_in[6];
    const float* Wv_b = (const float*)d_in[7];
    float* out = (float*)d_out;

    // Workspace layout (~9 MB)
    char*  ws    = (char*)d_ws;
    int*   sqlen = (int*)ws;                         // B ints
    float* last  = (float*)(ws + 8192);              // B*D
    float* Q     = last + (size_t)B * D;             // B*O
    float* T     = Q    + (size_t)B * O;             // B*D
    float* attn  = T    + (size_t)B * D;             // B*S
    float* xbar  = attn + (size_t)B * S;             // B*D

    prep_kernel<<<B, 256, 0, stream>>>(x, mask, sqlen, last, S, D);

    // Q = Last @ Wq^T + bq      (M=B, N=O, K=D; W stored [O,D] -> TRANSB)
    {
        const int waves = (B / 16) * (O / 64);
        gemm_wmma_f32<true, true, false><<<waves / 4, 128, 0, stream>>>(
            last, Wq_w, Wq_b, Q, B, O, D);
    }
    // T = Q @ Wk                (M=B, N=D, K=O; W stored [O,D] = [K,N] -> no trans)
    {
        const int waves = (B / 16) * (D / 64);
        gemm_wmma_f32<false, false, false><<<waves / 4, 128, 0, stream>>>(
            Q, Wk_w, nullptr, T, B, D, O);
    }

    scores_softmax_kernel<<<B, 256, 0, stream>>>(x, T, sqlen, attn, S, D);
    xbar_kernel<<<B, 256, 0, stream>>>(x, attn, sqlen, xbar, S, D);

    // out = tanh(xbar @ Wv^T + bv)
    {
        const int waves = (B / 16) * (O / 64);
        gemm_wmma_f32<true, true, true><<<waves / 4, 128, 0, stream>>>(
            xbar, Wv_w, Wv_b, out, B, O, D);
    }
}